// GaussianBlurLayer_48550310314146
// MI455X (gfx1250) — compile-verified
//
#include <hip/hip_runtime.h>
#include <hip/hip_bf16.h>

// Separable 31x31 (effective 25-tap) Gaussian blur, reflect padding,
// fp32 end-to-end, arithmetic via V_WMMA_F32_16X16X4_F32 banded matmuls.
// v2: SGPR-uniform wave id (no exec-masked WMMA), column-major patch +
// even strides so every WMMA operand pair is a single aligned ds_load_b64,
// operand preload so the 10-deep WMMA accumulate chain issues back-to-back.

typedef __attribute__((ext_vector_type(2))) float v2f;
typedef __attribute__((ext_vector_type(8))) float v8f;

#define IMG_H 512
#define IMG_W 512
#define TILE_H 16
#define TILE_W 128
#define PATCH_H 40           // TILE_H + 24 halo rows
#define PATCH_W 160          // TILE_W + 24 halo cols, padded to 10 chunks of 16
#define PCSTR 42             // column-major patch: stride between columns
                             // (even -> 8B-aligned b64 pairs; conflict-free)
#define TSTR 162             // row-major tmp stride (even, conflict-free)

__device__ __forceinline__ int reflect511(int i) {
    i = (i < 0) ? -i : i;          // reflect (no edge repeat), pad<=15 < H
    i = (i > 511) ? (1022 - i) : i;
    return i;
}

__global__ __launch_bounds__(256)
void gauss31_sep_wmma_f32(const float* __restrict__ x,
                          const float* __restrict__ wgt,
                          float* __restrict__ out) {
    __shared__ float sPatchT[PATCH_W * PCSTR];  // [col][row], 26.9 KB
    __shared__ float sTmp[TILE_H * TSTR];       // [row][col], 10.4 KB
    __shared__ float sGv[25];
    __shared__ float sGh[25];

    const int tid  = threadIdx.x;
    const int lane = tid & 31;
    // SGPR-uniform wave id: keeps chunk loops scalar so EXEC stays all-ones
    // around every WMMA (ISA requirement) and kills the exec-mask overhead.
    const int wave = __builtin_amdgcn_readfirstlane(tid >> 5);
    const int half = lane >> 4;         // lane group (0: lanes 0-15, 1: 16-31)
    const int n    = lane & 15;         // M (A) / N (B) / N (C,D) index

    const int plane = blockIdx.z;       // n*3 + c over (32,3) planes
    const int ch    = plane % 3;
    const int row0  = blockIdx.y * TILE_H;
    const int col0  = blockIdx.x * TILE_W;

    const float* __restrict__ xp = x + (size_t)plane * (IMG_H * IMG_W);
    const float* __restrict__ wp = wgt + ch * 31 * 31;

    // Exact 1D factors from the outer-product weight (no sqrt):
    //   gv[i]*gh[j] == w2d[i+3][j+3]
    if (tid < 25) {
        sGv[tid] = wp[(tid + 3) * 31 + 15];
        sGh[tid] = wp[15 * 31 + (tid + 3)] / wp[15 * 31 + 15];
    }

    // Cooperative reflected patch load, stored column-major.
    // Consecutive tid -> consecutive column -> LDS bank = 42*c mod 64,
    // distinct for 32 consecutive lanes (gcd-argument) -> conflict-free.
    for (int idx = tid; idx < PATCH_H * PATCH_W; idx += 256) {
        const int pr = idx / PATCH_W;
        const int pc = idx - pr * PATCH_W;
        const int gr = reflect511(row0 - 12 + pr);
        const int gc = reflect511(col0 - 12 + pc);
        sPatchT[pc * PCSTR + pr] = xp[gr * IMG_W + gc];
    }
    __syncthreads();

    // Per-lane banded-Toeplitz weight operands (K = 4*kk + 2*half + jj):
    //   vertical A:   Wv[M=n][j] = gv[j-n] if 0<=j-n<=24
    //   horizontal B: Wh[j][N=n] = gh[j-n] if 0<=j-n<=24
    v2f av[10], bh[10];
#pragma unroll
    for (int kk = 0; kk < 10; ++kk) {
#pragma unroll
        for (int jj = 0; jj < 2; ++jj) {
            const int j = 4 * kk + 2 * half + jj;
            const int d = j - n;
            const bool in = (d >= 0) && (d <= 24);
            av[kk][jj] = in ? sGv[d] : 0.0f;
            bh[kk][jj] = in ? sGh[d] : 0.0f;
        }
    }

    // -------- Pass 1 (vertical): tmp(16 x 160) = Wv(16x40) x patch(40x160) --
    // 10 column chunks of 16; wave w takes chunks w and w+8 (scalar loop).
    for (int q = wave; q < 10; q += 8) {
        const int c0 = q * 16;
        // Preload all 10 B pairs: rows (jb, jb+1) of column c0+n are
        // contiguous in the transposed patch -> one aligned ds_load_b64 each.
        const float* colp = &sPatchT[(c0 + n) * PCSTR + 2 * half];
        v2f b[10];
#pragma unroll
        for (int kk = 0; kk < 10; ++kk)
            b[kk] = *(const v2f*)(colp + 4 * kk);
        v8f acc = {};
#pragma unroll
        for (int kk = 0; kk < 10; ++kk)
            acc = __builtin_amdgcn_wmma_f32_16x16x4_f32(
                false, av[kk], false, b[kk], (short)0, acc, false, false);
#pragma unroll
        for (int i = 0; i < 8; ++i)                // D row = i + 8*half
            sTmp[(i + 8 * half) * TSTR + c0 + n] = acc[i];
    }
    __syncthreads();

    // -------- Pass 2 (horizontal): out(16x16) = tmp(16x[j0..j0+40)) x Wh ----
    {
        const int j0 = wave * 16;                  // 8 output chunks of 16
        // Preload all 10 A pairs: tmp[n][jb..jb+1] contiguous -> b64 each.
        const float* rowp = &sTmp[n * TSTR + j0 + 2 * half];
        v2f a[10];
#pragma unroll
        for (int kk = 0; kk < 10; ++kk)
            a[kk] = *(const v2f*)(rowp + 4 * kk);
        v8f acc = {};
#pragma unroll
        for (int kk = 0; kk < 10; ++kk)
            acc = __builtin_amdgcn_wmma_f32_16x16x4_f32(
                false, a[kk], false, bh[kk], (short)0, acc, false, false);
        float* __restrict__ op = out + (size_t)plane * (IMG_H * IMG_W);
#pragma unroll
        for (int i = 0; i < 8; ++i)
            op[(row0 + i + 8 * half) * IMG_W + col0 + j0 + n] = acc[i];
    }
}

extern "C" void kernel_launch(void* const* d_in, const int* in_sizes, int n_in,
                              void* d_out, int out_size, void* d_ws, size_t ws_size,
                              hipStream_t stream) {
    (void)in_sizes; (void)n_in; (void)d_ws; (void)ws_size; (void)out_size;
    const float* x   = (const float*)d_in[0];   // (32,3,512,512) fp32
    const float* wgt = (const float*)d_in[1];   // (3,1,31,31)   fp32
    float* out = (float*)d_out;                 // (32,3,512,512) fp32

    dim3 grid(IMG_W / TILE_W, IMG_H / TILE_H, 32 * 3);  // (4, 32, 96)
    dim3 block(256);                                    // 8 wave32 waves
    gauss31_sep_wmma_f32<<<grid, block, 0, stream>>>(x, wgt, out);
}